// EnhancedGatedTemporalFusion_22763326669310
// MI455X (gfx1250) — compile-verified
//
#include <hip/hip_runtime.h>

// ----------------------------------------------------------------------------
// EnhancedGatedTemporalFusion on MI455X (gfx1250, wave32).
//
// Exactness argument for truncation: reference weights = softmax over
// [S, S-1, ..., 1] in fp32 -> exp(-s) after max-subtraction, which is EXACTLY
// 0.0f for s >= ~104 (fp32 underflow). outputs[s] is bounded (sigmoid/tanh
// combinations), so all terms beyond s ~ 104 contribute exactly zero to the
// einsum. We compute T = 256 steps (2.5x margin). The attention branch is
// dead code in the reference and is eliminated.
//
// Matrix cores: V_WMMA_F32_16X16X4_F32 (fp32 in / fp32 acc) for the four
// (T x 300) x (300 x 1024) gate GEMMs -> no precision loss vs reference.
// ----------------------------------------------------------------------------

#define SEQ_T   256
#define IN_DIM  100
#define HID     1024
#define P1      300
#define OUTD    2613

typedef float v2f __attribute__((ext_vector_type(2)));
typedef float v8f __attribute__((ext_vector_type(8)));

// ---------------------------------------------------------------------------
// Kernel 1: x[s,i] = time_steps[s,i] + relu(pos_s*W_pe1[h]+b_pe1[h]) @ W_pe2[i,:]ᵀ + b_pe2[i]
// (T*IN_DIM threads, loop over h = 1024)
// ---------------------------------------------------------------------------
__global__ void k_posenc(const float* __restrict__ ts, const int* __restrict__ positions,
                         const float* __restrict__ Wpe1, const float* __restrict__ bpe1,
                         const float* __restrict__ Wpe2, const float* __restrict__ bpe2,
                         float* __restrict__ x)
{
    int idx = blockIdx.x * blockDim.x + threadIdx.x;
    if (idx >= SEQ_T * IN_DIM) return;
    int s = idx / IN_DIM;
    int i = idx - s * IN_DIM;
    float pos = (float)positions[s];
    float acc = bpe2[i];
    const float* w2 = Wpe2 + i * HID;   // W_pe2 is (IN_DIM, HID) row-major
    for (int h = 0; h < HID; ++h) {
        float pe = fmaf(pos, Wpe1[h], bpe1[h]);
        pe = fmaxf(pe, 0.0f);
        acc = fmaf(pe, w2[h], acc);
    }
    x[idx] = ts[s * IN_DIM + i] + acc;
}

// ---------------------------------------------------------------------------
// Kernel 2: combined[s,j] = x[s,:] . W_p1[j,:] + b_p1[j]   (W_p1 is (300,100))
// ---------------------------------------------------------------------------
__global__ void k_proj1(const float* __restrict__ x, const float* __restrict__ Wp1,
                        const float* __restrict__ bp1, float* __restrict__ combined)
{
    int idx = blockIdx.x * blockDim.x + threadIdx.x;
    if (idx >= SEQ_T * P1) return;
    int s = idx / P1;
    int j = idx - s * P1;
    const float* xr = x + s * IN_DIM;
    const float* wr = Wp1 + j * IN_DIM;
    float acc = bp1[j];
    #pragma unroll 4
    for (int k = 0; k < IN_DIM; ++k) acc = fmaf(xr[k], wr[k], acc);
    combined[idx] = acc;
}

// ---------------------------------------------------------------------------
// Kernel 3: gate GEMMs via V_WMMA_F32_16X16X4_F32.
//   pre[s,n] = combined[s,:] . W[n,:] + b[n]   (W is (1024,300) row-major)
//   gate 0/2 -> sigmoid -> G1/G2 ; gate 1/3 -> tanh -> C1/C2
// One wave per 16x16 output tile; blockDim=(32,8); grid=(HID/128, T/16, 4).
// Fragment layouts per CDNA5 ISA §7.12.2 (f32, 16x16x4):
//   A: lanes 0-15 hold M=lane, K=k0..k0+1 in 2 VGPRs; lanes 16-31 hold K=k0+2..3
//   B: lanes 0-15 hold N=lane, K=k0..k0+1 ;            lanes 16-31 hold K=k0+2..3
//   C/D: VGPR r -> row m0 + r + 8*(lane>=16), col n0 + (lane&15)
// ---------------------------------------------------------------------------
__global__ void k_gates_wmma(const float* __restrict__ combined,
    const float* __restrict__ Wg1, const float* __restrict__ bg1, float* __restrict__ G1,
    const float* __restrict__ Wc1, const float* __restrict__ bc1, float* __restrict__ C1,
    const float* __restrict__ Wg2, const float* __restrict__ bg2, float* __restrict__ G2,
    const float* __restrict__ Wc2, const float* __restrict__ bc2, float* __restrict__ C2)
{
    const int gate = blockIdx.z;
    const float* W; const float* bias; float* dst;
    switch (gate) {
        case 0:  W = Wg1; bias = bg1; dst = G1; break;
        case 1:  W = Wc1; bias = bc1; dst = C1; break;
        case 2:  W = Wg2; bias = bg2; dst = G2; break;
        default: W = Wc2; bias = bc2; dst = C2; break;
    }
    const int lane = threadIdx.x;                       // 0..31
    const int n0 = (blockIdx.x * blockDim.y + threadIdx.y) * 16;
    const int m0 = blockIdx.y * 16;
    const int lo = lane & 15;
    const int hi = lane >> 4;                           // 0 or 1 -> K subgroup

    // 8-byte aligned: stride 300 floats is even, k multiple of 4, 2*hi even.
    const float* arow = combined + (m0 + lo) * P1 + 2 * hi;
    const float* brow = W        + (n0 + lo) * P1 + 2 * hi;

    v8f acc = {};
    #pragma unroll 5
    for (int k = 0; k < P1; k += 4) {                   // 75 WMMA issues
        float2 a2 = *(const float2*)(arow + k);
        float2 b2 = *(const float2*)(brow + k);
        v2f av; av.x = a2.x; av.y = a2.y;
        v2f bv; bv.x = b2.x; bv.y = b2.y;
        acc = __builtin_amdgcn_wmma_f32_16x16x4_f32(
                  /*neg_a=*/false, av, /*neg_b=*/false, bv,
                  /*c_mod=*/(short)0, acc, /*reuse_a=*/false, /*reuse_b=*/false);
    }

    const int  ncol   = n0 + lo;
    const int  rbase  = m0 + hi * 8;
    const float bn    = bias[ncol];
    const bool is_sig = (gate & 1) == 0;
    #pragma unroll
    for (int r = 0; r < 8; ++r) {
        float v = acc[r] + bn;
        v = is_sig ? (1.0f / (1.0f + expf(-v))) : tanhf(v);
        dst[(rbase + r) * HID + ncol] = v;
    }
}

// ---------------------------------------------------------------------------
// Kernel 4: sequential scan over T steps fused with exp-weighted reduction.
// One block, 1024 threads, one lane per hidden channel.
//   h1 = g1*h1 + (1-g1)*c1 = c1 + g1*(h1-c1)  (same algebra)
//   weighted[h] = sum_s (h1+h2)*exp(-s) / Z,  Z = sum_n exp(-n)
// ---------------------------------------------------------------------------
__global__ void k_scan(const float* __restrict__ G1, const float* __restrict__ C1,
                       const float* __restrict__ G2, const float* __restrict__ C2,
                       float* __restrict__ weighted)
{
    int h = threadIdx.x;
    float h1 = 0.0f, h2 = 0.0f, acc = 0.0f;
    for (int s = 0; s < SEQ_T; ++s) {
        int o = s * HID + h;
        float g1 = G1[o], c1 = C1[o], g2 = G2[o], c2 = C2[o];
        h1 = fmaf(g1, h1 - c1, c1);
        h2 = fmaf(g2, h2 - c2, c2);
        acc = fmaf(h1 + h2, expf(-(float)s), acc);
    }
    float Z = 0.0f;
    for (int n = 0; n < SEQ_T; ++n) Z += expf(-(float)n);   // == softmax denom in fp32
    weighted[h] = acc / Z;
}

// ---------------------------------------------------------------------------
// Kernel 5: out[o] = weighted . W_out[o,:] + b_out[o]   (wave32 per output)
// ---------------------------------------------------------------------------
__global__ void k_outproj(const float* __restrict__ weighted, const float* __restrict__ Wout,
                          const float* __restrict__ bout, float* __restrict__ out)
{
    int gtid = blockIdx.x * blockDim.x + threadIdx.x;
    int o    = gtid >> 5;
    int lane = threadIdx.x & 31;
    if (o >= OUTD) return;
    const float* wr = Wout + o * HID;
    float acc = 0.0f;
    #pragma unroll 4
    for (int i = lane; i < HID; i += 32) acc = fmaf(wr[i], weighted[i], acc);
    #pragma unroll
    for (int off = 16; off > 0; off >>= 1) acc += __shfl_xor(acc, off, 32);
    if (lane == 0) out[o] = acc + bout[o];
}

// ---------------------------------------------------------------------------
extern "C" void kernel_launch(void* const* d_in, const int* in_sizes, int n_in,
                              void* d_out, int out_size, void* d_ws, size_t ws_size,
                              hipStream_t stream)
{
    const float* ts    = (const float*)d_in[0];
    // d_in[1] global_code: dead (attention output unused)
    const int*   pos   = (const int*)d_in[2];
    const float* Wpe1  = (const float*)d_in[3];
    const float* bpe1  = (const float*)d_in[4];
    const float* Wpe2  = (const float*)d_in[5];
    const float* bpe2  = (const float*)d_in[6];
    // d_in[7..10] attention params: dead
    const float* Wp1   = (const float*)d_in[11];
    const float* bp1   = (const float*)d_in[12];
    const float* Wg1   = (const float*)d_in[13];
    const float* bg1   = (const float*)d_in[14];
    const float* Wc1   = (const float*)d_in[15];
    const float* bc1   = (const float*)d_in[16];
    const float* Wg2   = (const float*)d_in[17];
    const float* bg2   = (const float*)d_in[18];
    const float* Wc2   = (const float*)d_in[19];
    const float* bc2   = (const float*)d_in[20];
    const float* Wout  = (const float*)d_in[21];
    const float* bout  = (const float*)d_in[22];
    float* out = (float*)d_out;

    // Workspace layout (floats). Total ~4.6 MB.
    float* ws       = (float*)d_ws;
    float* x        = ws;                                  // 256*100
    float* combined = x + SEQ_T * IN_DIM;                  // 256*300
    float* G1       = combined + SEQ_T * P1;               // 256*1024
    float* C1       = G1 + SEQ_T * HID;
    float* G2       = C1 + SEQ_T * HID;
    float* C2       = G2 + SEQ_T * HID;
    float* weighted = C2 + SEQ_T * HID;                    // 1024

    // 1) positional MLP fold -> x
    {
        int n = SEQ_T * IN_DIM;
        k_posenc<<<(n + 255) / 256, 256, 0, stream>>>(ts, pos, Wpe1, bpe1, Wpe2, bpe2, x);
    }
    // 2) combined = x @ W_p1^T + b_p1
    {
        int n = SEQ_T * P1;
        k_proj1<<<(n + 255) / 256, 256, 0, stream>>>(x, Wp1, bp1, combined);
    }
    // 3) four gate GEMMs via fp32 WMMA; grid.z selects gate+activation
    {
        dim3 blk(32, 8, 1);                // 8 waves/block, one 16x16 tile each
        dim3 grd(HID / (16 * 8), SEQ_T / 16, 4);
        k_gates_wmma<<<grd, blk, 0, stream>>>(combined,
                                              Wg1, bg1, G1, Wc1, bc1, C1,
                                              Wg2, bg2, G2, Wc2, bc2, C2);
    }
    // 4) scan + exp-weighted reduction
    k_scan<<<1, HID, 0, stream>>>(G1, C1, G2, C2, weighted);
    // 5) output projection (wave per output row)
    {
        int waves  = OUTD;
        int blocks = (waves + 7) / 8;      // 8 waves (256 threads) per block
        k_outproj<<<blocks, 256, 0, stream>>>(weighted, Wout, bout, out);
    }
}